// RGCNLayer_1434519077565
// MI455X (gfx1250) — compile-verified
//
#include <hip/hip_runtime.h>
#include <cstdint>

// Problem constants from the reference (needed at compile time for LDS sizing).
#define RANK   64
#define NREL   500
#define REL_FLOATS (NREL * RANK)        // 32000 floats
#define REL_BYTES  (REL_FLOATS * 4)     // 128000 bytes -> fits in 320KB CDNA5 LDS

__global__ void rgcn_zero_kernel(float4* __restrict__ out, int n4) {
    for (int i = blockIdx.x * blockDim.x + threadIdx.x; i < n4;
         i += gridDim.x * blockDim.x) {
        out[i] = make_float4(0.f, 0.f, 0.f, 0.f);
    }
}

__global__ __launch_bounds__(256)
void rgcn_scatter_kernel(const float* __restrict__ h,
                         const float* __restrict__ rel,
                         const float* __restrict__ norm,
                         const int*   __restrict__ src,
                         const int*   __restrict__ dst,
                         const int*   __restrict__ etype,
                         float*       __restrict__ out,
                         int n_edges) {
    // Whole relation table lives in LDS (128 KB of the 320 KB CDNA5 WGP LDS).
    __shared__ float lds_rel[REL_FLOATS];

    // ---- Stage rel -> LDS with gfx1250 async global->LDS b128 copies ----
    {
        const int n16 = REL_BYTES / 16; // 8000 x 16B chunks
        for (int i = threadIdx.x; i < n16; i += blockDim.x) {
            // Low 32 bits of a generic pointer to LDS are the LDS byte offset.
            unsigned lds_off = (unsigned)(uintptr_t)(const void*)(&lds_rel[i * 4]);
            unsigned long long ga =
                (unsigned long long)(uintptr_t)rel + (unsigned long long)i * 16ull;
            asm volatile("global_load_async_to_lds_b128 %0, %1, off"
                         :
                         : "v"(lds_off), "v"(ga)
                         : "memory");
        }
        asm volatile("s_wait_asynccnt 0x0" ::: "memory");
        __syncthreads();
    }

    // ---- Each wave32 owns contiguous 32-edge chunks; lane l owns features {2l,2l+1}.
    const int lane    = threadIdx.x & 31;
    const int wave    = blockIdx.x * (blockDim.x >> 5) + (threadIdx.x >> 5);
    const int n_waves = gridDim.x * (blockDim.x >> 5);

    for (int base = wave * 32; base < n_edges; base += n_waves * 32) {
        // Lane-batched index loads: one coalesced b32 load per edge (not 32x).
        const int e = base + lane;
        int   s = 0, d = 0, t = 0;
        float nm = 0.f;
        if (e < n_edges) {
            s  = src[e];
            d  = dst[e];
            t  = etype[e];
            nm = norm[d];
            // Warm L2/WGP$ for this chunk's 32 h-rows (global_prefetch_b8).
            __builtin_prefetch(h + (size_t)s * RANK, 0, 0);
        }

        const int cnt = (n_edges - base < 32) ? (n_edges - base) : 32; // wave-uniform

#pragma unroll
        for (int j = 0; j < 32; ++j) {
            if (j >= cnt) break;                       // uniform break: EXEC stays full
            const int   sj = __shfl(s, j);             // v_readlane_b32 (immediate lane)
            const int   dj = __shfl(d, j);
            const int   tj = __shfl(t, j);
            const float nj = __shfl(nm, j);

            const float2 hv = ((const float2*)(h + (size_t)sj * RANK))[lane]; // 256B coalesced
            const float2 rv = ((const float2*)(lds_rel + tj * RANK))[lane];   // ds_load_b64

            float* o = out + (size_t)dj * RANK + lane * 2;
            unsafeAtomicAdd(o,     (hv.x + rv.x) * nj);   // global_atomic_add_f32
            unsafeAtomicAdd(o + 1, (hv.y + rv.y) * nj);
        }
    }
}

extern "C" void kernel_launch(void* const* d_in, const int* in_sizes, int n_in,
                              void* d_out, int out_size, void* d_ws, size_t ws_size,
                              hipStream_t stream) {
    const float* h     = (const float*)d_in[0];
    const float* rel   = (const float*)d_in[1];
    const float* norm  = (const float*)d_in[2];
    const int*   src   = (const int*)d_in[3];
    const int*   dst   = (const int*)d_in[4];
    const int*   etype = (const int*)d_in[5];
    float*       out   = (float*)d_out;

    const int n_edges = in_sizes[3];

    // Zero the (poisoned) output accumulator.
    const int n4 = out_size / 4;
    rgcn_zero_kernel<<<(n4 + 255) / 256, 256, 0, stream>>>((float4*)out, n4);

    // 512 blocks x 256 threads (8 wave32s each): 2 blocks/WGP at 128KB LDS,
    // ~1560 edges per block amortizes the 128KB LDS staging ~3x in rel reads.
    rgcn_scatter_kernel<<<512, 256, 0, stream>>>(h, rel, norm, src, dst, etype,
                                                 out, n_edges);
}